// CRFLayer_44083544326968
// MI455X (gfx1250) — compile-verified
//
#include <hip/hip_runtime.h>
#include <stdint.h>

#ifndef __has_builtin
#define __has_builtin(x) 0
#endif

constexpr int CB = 256;  // batch
constexpr int CT = 512;  // time
constexpr int CL = 128;  // labels

typedef unsigned int v4u __attribute__((ext_vector_type(4)));
typedef int          v8i __attribute__((ext_vector_type(8)));
typedef int          v4i __attribute__((ext_vector_type(4)));

#define HAVE_TDM (__has_builtin(__builtin_amdgcn_tensor_load_to_lds) && \
                  __has_builtin(__builtin_amdgcn_s_wait_tensorcnt))

// DMA the 128x128 f32 transition matrix (64 KB) global -> LDS via the Tensor
// Data Mover. s must sit at dynamic-LDS offset 0 (no static LDS in callers).
__device__ __forceinline__ void load_trans_to_lds(const float* __restrict__ g,
                                                  float* __restrict__ s, int lane) {
#if HAVE_TDM
  unsigned long long ga = (unsigned long long)(uintptr_t)g;
  v4u g0;
  g0.x = 1u;                                                  // count=1 (user D#)
  g0.y = 0u;                                                  // lds_addr = 0
  g0.z = (unsigned)(ga & 0xffffffffu);                        // global_addr[31:0]
  g0.w = (unsigned)((ga >> 32) & 0x01ffffffu) | 0x80000000u;  // addr[56:32] | type=2
  v8i g1;
  g1[0] = 0x00020000;   // data_size=2 (4B); no multicast/iterate/pad
  g1[1] = (CL << 16);   // tensor_dim0[15:0]=128 at bits [63:48]
  g1[2] = (CL << 16);   // tensor_dim0 hi=0 | tensor_dim1[15:0]=128
  g1[3] = (CL << 16);   // tensor_dim1 hi=0 | tile_dim0=128
  g1[4] = CL;           // tile_dim1=128, tile_dim2=0
  g1[5] = CL;           // tensor_dim0_stride[31:0]=128
  g1[6] = 0;            // stride0 hi | stride1 lo (unused for 2D)
  g1[7] = 0;
  v4i z4 = {0, 0, 0, 0};
#if __clang_major__ >= 23
  v8i z8 = {0, 0, 0, 0, 0, 0, 0, 0};
  __builtin_amdgcn_tensor_load_to_lds(g0, g1, z4, z4, z8, 0);
#else
  __builtin_amdgcn_tensor_load_to_lds(g0, g1, z4, z4, 0);
#endif
  __builtin_amdgcn_s_wait_tensorcnt(0);
  (void)s; (void)lane;
#else
#pragma unroll 4
  for (int idx = lane * 4; idx < CL * CL; idx += 32 * 4)
    *(float4*)(s + idx) = *(const float4*)(g + idx);
#endif
  __syncthreads();
}

// First-max argmax over 128 values (4 per lane, i = 4*lane + k), matching
// jnp.argmax tie-breaking: strict '>' ascending locally, lower index wins ties
// across lanes.
__device__ __forceinline__ void wave_argmax(float4 v, int lane, float& bv, int& bi) {
  bv = v.x; bi = 4 * lane;
  if (v.y > bv) { bv = v.y; bi = 4 * lane + 1; }
  if (v.z > bv) { bv = v.z; bi = 4 * lane + 2; }
  if (v.w > bv) { bv = v.w; bi = 4 * lane + 3; }
#pragma unroll
  for (int off = 16; off >= 1; off >>= 1) {
    float ov = __shfl_xor(bv, off, 32);
    int   oi = __shfl_xor(bi, off, 32);
    if (ov > bv || (ov == bv && oi < bi)) { bv = ov; bi = oi; }
  }
}

// ---------------- Path A: store carry states, recompute argmax backward ------

__global__ __launch_bounds__(32)
void crf_fwd_state(const float* __restrict__ score, const int* __restrict__ seqlen,
                   const float* __restrict__ trans, float* __restrict__ S) {
  extern __shared__ float smem[];
  float* s_trans = smem;            // 16384 floats @ LDS offset 0
  float* s_state = smem + CL * CL;  // 128 floats
  const int lane = threadIdx.x;
  const int b = blockIdx.x;
  load_trans_to_lds(trans, s_trans, lane);

  const float* sc = score + (size_t)b * CT * CL;
  float4 carry = ((const float4*)sc)[lane];
  ((float4*)s_state)[lane] = carry;
  ((float4*)(S + (size_t)b * CL))[lane] = carry;  // S[0] = score[:,0,:]
  __syncthreads();

  const int sl = seqlen[b];
  const int tend = sl < CT ? sl : CT;  // steps t >= seq_len are no-ops
  const float NEG = -3.402823466e38f;
  for (int t = 1; t < tend; ++t) {
    const float* em = sc + (size_t)t * CL;
    if (t + 1 < tend) __builtin_prefetch(em + CL + 4 * lane, 0, 3);
    float4 best = make_float4(NEG, NEG, NEG, NEG);
#define MP_STEP(ii, sv)                                              \
  { const float4 r = ((const float4*)(s_trans + (ii) * CL))[lane];   \
    best.x = fmaxf(best.x, (sv) + r.x);                              \
    best.y = fmaxf(best.y, (sv) + r.y);                              \
    best.z = fmaxf(best.z, (sv) + r.z);                              \
    best.w = fmaxf(best.w, (sv) + r.w); }
#pragma unroll
    for (int i = 0; i < CL; i += 4) {
      const float4 sb = ((const float4*)s_state)[i >> 2];  // uniform broadcast
      MP_STEP(i + 0, sb.x)
      MP_STEP(i + 1, sb.y)
      MP_STEP(i + 2, sb.z)
      MP_STEP(i + 3, sb.w)
    }
#undef MP_STEP
    const float4 em4 = ((const float4*)em)[lane];
    carry.x = best.x + em4.x;
    carry.y = best.y + em4.y;
    carry.z = best.z + em4.z;
    carry.w = best.w + em4.w;
    ((float4*)s_state)[lane] = carry;
    ((float4*)(S + ((size_t)t * CB + b) * CL))[lane] = carry;
    __syncthreads();
  }
}

__global__ __launch_bounds__(32)
void crf_bt_state(const int* __restrict__ seqlen, const float* __restrict__ trans,
                  const float* __restrict__ S, int* __restrict__ tags,
                  float* __restrict__ bscore) {
  extern __shared__ float s_tt[];  // transposed transitions: s_tt[j*128 + i]
  const int lane = threadIdx.x;
  const int b = blockIdx.x;
  for (int idx = lane; idx < CL * CL; idx += 32) {
    const int i = idx >> 7, j = idx & (CL - 1);
    s_tt[j * CL + i] = trans[idx];
  }
  __syncthreads();

  const int sl = seqlen[b];
  // final carry = S[seq_len-1] (later steps never modified it)
  const float4 fv = ((const float4*)(S + ((size_t)(sl - 1) * CB + b) * CL))[lane];
  float bv; int tag;
  wave_argmax(fv, lane, bv, tag);
  if (lane == 0) bscore[b] = bv;

  int* tout = tags + (size_t)b * CT;
  for (int t = sl + lane; t < CT; t += 32) tout[t] = tag;  // invalid region: last_tag
  for (int t = sl - 1; t >= 1; --t) {
    if (lane == 0) tout[t] = tag;
    const float4 s4 = ((const float4*)(S + ((size_t)(t - 1) * CB + b) * CL))[lane];
    if (t > 3) __builtin_prefetch(S + ((size_t)(t - 4) * CB + b) * CL + 4 * lane, 0, 3);
    const float4 tr = ((const float4*)(s_tt + (size_t)tag * CL))[lane];
    const float4 c = make_float4(s4.x + tr.x, s4.y + tr.y, s4.z + tr.z, s4.w + tr.w);
    wave_argmax(c, lane, bv, tag);  // recomputed bp[t][tag], states match bitwise
  }
  if (lane == 0) tout[0] = tag;
}

// ---------------- Path B (small workspace): byte backpointers ----------------

__global__ __launch_bounds__(32)
void crf_fwd_bp(const float* __restrict__ score, const int* __restrict__ seqlen,
                const float* __restrict__ trans, unsigned char* __restrict__ bp,
                float* __restrict__ fst) {
  extern __shared__ float smem[];
  float* s_trans = smem;
  float* s_state = smem + CL * CL;
  const int lane = threadIdx.x;
  const int b = blockIdx.x;
  load_trans_to_lds(trans, s_trans, lane);

  const float* sc = score + (size_t)b * CT * CL;
  float4 carry = ((const float4*)sc)[lane];
  ((float4*)s_state)[lane] = carry;
  __syncthreads();

  const int sl = seqlen[b];
  const int tend = sl < CT ? sl : CT;
  const float NEG = -3.402823466e38f;
  for (int t = 1; t < tend; ++t) {
    const float* em = sc + (size_t)t * CL;
    if (t + 1 < tend) __builtin_prefetch(em + CL + 4 * lane, 0, 3);
    float4 best = make_float4(NEG, NEG, NEG, NEG);
    int b0 = 0, b1 = 0, b2 = 0, b3 = 0;
#define MPA_STEP(ii, sv)                                             \
  { const float4 r = ((const float4*)(s_trans + (ii) * CL))[lane];   \
    float c;                                                         \
    c = (sv) + r.x; if (c > best.x) { best.x = c; b0 = (ii); }       \
    c = (sv) + r.y; if (c > best.y) { best.y = c; b1 = (ii); }       \
    c = (sv) + r.z; if (c > best.z) { best.z = c; b2 = (ii); }       \
    c = (sv) + r.w; if (c > best.w) { best.w = c; b3 = (ii); } }
#pragma unroll
    for (int i = 0; i < CL; i += 4) {
      const float4 sb = ((const float4*)s_state)[i >> 2];
      MPA_STEP(i + 0, sb.x)
      MPA_STEP(i + 1, sb.y)
      MPA_STEP(i + 2, sb.z)
      MPA_STEP(i + 3, sb.w)
    }
#undef MPA_STEP
    const unsigned pb = (unsigned)b0 | ((unsigned)b1 << 8) |
                        ((unsigned)b2 << 16) | ((unsigned)b3 << 24);
    *(unsigned*)(bp + ((size_t)(t - 1) * CB + b) * CL + 4 * lane) = pb;
    const float4 em4 = ((const float4*)em)[lane];
    carry.x = best.x + em4.x;
    carry.y = best.y + em4.y;
    carry.z = best.z + em4.z;
    carry.w = best.w + em4.w;
    ((float4*)s_state)[lane] = carry;
    __syncthreads();
  }
  ((float4*)(fst + (size_t)b * CL))[lane] = carry;
}

__global__ __launch_bounds__(32)
void crf_bt_bp(const int* __restrict__ seqlen, const unsigned char* __restrict__ bp,
               const float* __restrict__ fst, int* __restrict__ tags,
               float* __restrict__ bscore) {
  const int lane = threadIdx.x;
  const int b = blockIdx.x;
  const int sl = seqlen[b];
  const float4 fv = ((const float4*)(fst + (size_t)b * CL))[lane];
  float bv; int tag;
  wave_argmax(fv, lane, bv, tag);
  if (lane == 0) bscore[b] = bv;
  int* tout = tags + (size_t)b * CT;
  for (int t = sl + lane; t < CT; t += 32) tout[t] = tag;
  if (lane == 0) {
    for (int t = sl - 1; t >= 1; --t) {
      tout[t] = tag;
      tag = bp[((size_t)(t - 1) * CB + b) * CL + tag];
    }
    tout[0] = tag;
  }
}

extern "C" void kernel_launch(void* const* d_in, const int* in_sizes, int n_in,
                              void* d_out, int out_size, void* d_ws, size_t ws_size,
                              hipStream_t stream) {
  (void)in_sizes; (void)n_in; (void)out_size;
  const float* score  = (const float*)d_in[0];   // [B,T,L] f32
  const int*   seqlen = (const int*)d_in[1];     // [B] i32
  const float* trans  = (const float*)d_in[2];   // [L,L] f32
  int*   tags   = (int*)d_out;                   // [B,T] i32
  float* bscore = (float*)d_out + (size_t)CB * CT;  // [B] f32

  const size_t needA    = sizeof(float) * (size_t)CT * CB * CL;  // 64 MiB state log
  const size_t smem_fwd = sizeof(float) * (CL * CL + CL);
  const size_t smem_bt  = sizeof(float) * CL * CL;

  if (ws_size >= needA) {
    float* S = (float*)d_ws;
    crf_fwd_state<<<CB, 32, smem_fwd, stream>>>(score, seqlen, trans, S);
    crf_bt_state<<<CB, 32, smem_bt, stream>>>(seqlen, trans, S, tags, bscore);
  } else {
    unsigned char* bp = (unsigned char*)d_ws;                       // (T-1)*B*L bytes
    float* fstv = (float*)((char*)d_ws + (size_t)(CT - 1) * CB * CL);
    crf_fwd_bp<<<CB, 32, smem_fwd, stream>>>(score, seqlen, trans, bp, fstv);
    crf_bt_bp<<<CB, 32, 0, stream>>>(seqlen, bp, fstv, tags, bscore);
  }
}